// GCN_19241453486477
// MI455X (gfx1250) — compile-verified
//
#include <hip/hip_runtime.h>
#include <hip/hip_bf16.h>

#define NNODES 50000
#define NEDGES 800000
#define C      128   // IN == OUT == 128

typedef __attribute__((ext_vector_type(2))) float v2f;
typedef __attribute__((ext_vector_type(8))) float v8f;

// ---------------------------------------------------------------------------
// zero: n must be a multiple of 4
__global__ __launch_bounds__(256) void gcn_zero_f32(float* __restrict__ p, int n4) {
    int i = blockIdx.x * blockDim.x + threadIdx.x;
    if (i < n4) {
        float4 z = make_float4(0.f, 0.f, 0.f, 0.f);
        *(float4*)(p + (size_t)i * 4) = z;
    }
}

// ---------------------------------------------------------------------------
// degree accumulation (float counts are exact up to 2^24)
__global__ __launch_bounds__(256) void gcn_degree(const int* __restrict__ src,
                                                  const int* __restrict__ dst,
                                                  float* __restrict__ degS,
                                                  float* __restrict__ degD,
                                                  int nedges) {
    int i = blockIdx.x * blockDim.x + threadIdx.x;
    if (i < nedges) {
        __hip_atomic_fetch_add(degS + src[i], 1.0f, __ATOMIC_RELAXED, __HIP_MEMORY_SCOPE_AGENT);
        __hip_atomic_fetch_add(degD + dst[i], 1.0f, __ATOMIC_RELAXED, __HIP_MEMORY_SCOPE_AGENT);
    }
}

// in-place deg -> clip(deg,1)^-0.5 over both arrays (contiguous, 2N elements)
__global__ __launch_bounds__(256) void gcn_norm(float* __restrict__ p, int n) {
    int i = blockIdx.x * blockDim.x + threadIdx.x;
    if (i < n) p[i] = rsqrtf(fmaxf(p[i], 1.0f));
}

// ---------------------------------------------------------------------------
// y[r,:] = (h[r,:] * ns[r]) @ W    via V_WMMA_F32_16X16X4_F32
// block: 256 threads = 8 waves, owns 128 rows; wave w owns 16-col tile w.
// A tile staged raw in LDS via GLOBAL_LOAD_ASYNC_TO_LDS_B128 (ASYNCcnt),
// ns-scaling applied at fragment read. B fragments preloaded to registers
// once per wave (64 VGPRs) and reused across all 8 row tiles.
__global__ __launch_bounds__(256) void gcn_gemm_scaled(const float* __restrict__ h,
                                                       const float* __restrict__ ns,
                                                       const float* __restrict__ W,
                                                       float* __restrict__ y,
                                                       int nrows) {
    // Row stride 132 -> bank = (4*M + K) % 64: conflict-free fragment reads
    // for both half-waves.
    __shared__ float Al[128][132];

    const int tid  = threadIdx.x;
    const int row0 = blockIdx.x * 128;

    // ---- async stage of the raw 128x128 A tile into LDS -------------------
    {
        const unsigned lbase = (unsigned)(size_t)(&Al[0][0]);
        for (int i = tid; i < 128 * 32; i += 256) {
            int r  = i >> 5;            // local row
            int c4 = (i & 31) << 2;     // column (multiple of 4)
            int gr = row0 + r;
            if (gr < nrows) {           // OOB lanes EXEC-masked off the copy
                unsigned     laddr = lbase + (unsigned)(r * 132 + c4) * 4u;
                const float* gaddr = h + (size_t)gr * C + c4;
                asm volatile("global_load_async_to_lds_b128 %0, %1, off"
                             :: "v"(laddr), "v"(gaddr)
                             : "memory");
            }
        }
        asm volatile("s_wait_asynccnt 0x0" ::: "memory");
    }
    __syncthreads();

    const int  wave = tid >> 5;          // 0..7 -> column tile
    const int  lane = tid & 31;
    const int  lh   = lane & 15;
    const bool hi   = lane >= 16;        // half-wave holds K+2/K+3
    const int  ct   = wave * 16;

    // ---- preload all B fragments for this column tile (reused 8x) ---------
    v2f bfrag[32];
#pragma unroll
    for (int kk = 0; kk < 32; ++kk) {
        const int kb = (kk << 2) + (hi ? 2 : 0);
        bfrag[kk].x = W[(size_t)kb * C + ct + lh];
        bfrag[kk].y = W[(size_t)(kb + 1) * C + ct + lh];
    }

    for (int rt = 0; rt < 8; ++rt) {
        const int   arow = rt * 16 + lh;
        const int   grow = row0 + arow;
        const float s    = (grow < nrows) ? ns[grow] : 0.f;  // also kills OOB garbage
        v8f acc = {};
#pragma unroll
        for (int kk = 0; kk < 32; ++kk) {
            const int ka = (kk << 2) + (hi ? 2 : 0);
            v2f a;
            a.x = Al[arow][ka] * s;
            a.y = Al[arow][ka + 1] * s;
            acc = __builtin_amdgcn_wmma_f32_16x16x4_f32(
                false, a, false, bfrag[kk], (short)0, acc, false, false);
        }
        // C/D layout: VGPR i -> M = i (lanes 0-15) / 8+i (lanes 16-31), N = lane%16
#pragma unroll
        for (int i = 0; i < 8; ++i) {
            int m  = hi ? (8 + i) : i;
            int gr = row0 + rt * 16 + m;
            if (gr < nrows) y[(size_t)gr * C + ct + lh] = acc[i];
        }
    }
}

// ---------------------------------------------------------------------------
// agg[dst[e],:] += y[src[e],:]   one wave per edge, one float4 per lane.
// Working set (y + agg = 51 MB) is L2-resident; relaxed agent-scope fp32
// atomics resolve at the L2 atomic units.
__global__ __launch_bounds__(256) void gcn_spmm_scatter(const float* __restrict__ y,
                                                        const int* __restrict__ src,
                                                        const int* __restrict__ dst,
                                                        float* __restrict__ agg,
                                                        int nedges) {
    int e    = blockIdx.x * 8 + (threadIdx.x >> 5);
    int lane = threadIdx.x & 31;
    if (e >= nedges) return;
    int s = src[e];
    int d = dst[e];
    const float4 v = *(const float4*)(y + (size_t)s * C + lane * 4);
    float* o = agg + (size_t)d * C + lane * 4;
    __hip_atomic_fetch_add(o + 0, v.x, __ATOMIC_RELAXED, __HIP_MEMORY_SCOPE_AGENT);
    __hip_atomic_fetch_add(o + 1, v.y, __ATOMIC_RELAXED, __HIP_MEMORY_SCOPE_AGENT);
    __hip_atomic_fetch_add(o + 2, v.z, __ATOMIC_RELAXED, __HIP_MEMORY_SCOPE_AGENT);
    __hip_atomic_fetch_add(o + 3, v.w, __ATOMIC_RELAXED, __HIP_MEMORY_SCOPE_AGENT);
}

// ---------------------------------------------------------------------------
// out[r,c] = prelu(agg[r,c]*nd[r]); hg[c] += sum_r out[r,c]  (block = 64 rows)
__global__ __launch_bounds__(256) void gcn_finalize(const float* __restrict__ agg,
                                                    const float* __restrict__ nd,
                                                    const float* __restrict__ slope,
                                                    float* __restrict__ out,
                                                    float* __restrict__ hg,
                                                    int nrows) {
    __shared__ float red[128];
    const int col  = threadIdx.x & 127;
    const int rsub = threadIdx.x >> 7;   // 0 or 1
    const int row0 = blockIdx.x * 64;
    const float a  = slope[0];

    float acc = 0.f;
    for (int r = rsub; r < 64; r += 2) {
        int row = row0 + r;
        if (row < nrows) {
            float v = agg[(size_t)row * C + col] * nd[row];
            v = (v >= 0.f) ? v : a * v;
            out[(size_t)row * C + col] = v;
            acc += v;
        }
    }
    if (rsub == 1) red[col] = acc;
    __syncthreads();
    if (rsub == 0) {
        __hip_atomic_fetch_add(hg + col, acc + red[col],
                               __ATOMIC_RELAXED, __HIP_MEMORY_SCOPE_AGENT);
    }
}

// ---------------------------------------------------------------------------
extern "C" void kernel_launch(void* const* d_in, const int* in_sizes, int n_in,
                              void* d_out, int out_size, void* d_ws, size_t ws_size,
                              hipStream_t stream) {
    const float* feat = (const float*)d_in[0];
    const int*   src  = (const int*)d_in[1];
    const int*   dst  = (const int*)d_in[2];
    const float* W0   = (const float*)d_in[3];
    const float* W1   = (const float*)d_in[4];
    const float* W2   = (const float*)d_in[5];
    const float* a0   = (const float*)d_in[6];
    const float* a1   = (const float*)d_in[7];
    const float* a2   = (const float*)d_in[8];

    float* out_h = (float*)d_out;              // N*C final features
    float* hg    = out_h + (size_t)NNODES * C; // 3*128 pooled readout

    float* ws    = (float*)d_ws;
    float* ns    = ws;                         // N
    float* nd    = ws + NNODES;                // N
    float* hcur  = ws + 2 * NNODES;            // N*C
    float* yproj = hcur + (size_t)NNODES * C;  // N*C
    float* agg   = yproj + (size_t)NNODES * C; // N*C

    const int NC = NNODES * C;

    // degrees -> symmetric sqrt norms
    gcn_zero_f32<<<(2 * NNODES / 4 + 255) / 256, 256, 0, stream>>>(ns, 2 * NNODES / 4);
    gcn_zero_f32<<<(384 / 4 + 255) / 256, 256, 0, stream>>>(hg, 384 / 4);
    gcn_degree<<<(NEDGES + 255) / 256, 256, 0, stream>>>(src, dst, ns, nd, NEDGES);
    gcn_norm<<<(2 * NNODES + 255) / 256, 256, 0, stream>>>(ns, 2 * NNODES);

    const int gemm_blocks = (NNODES + 127) / 128;
    const int spmm_blocks = (NEDGES + 7) / 8;
    const int fin_blocks  = (NNODES + 63) / 64;
    const int zero_blocks = (NC / 4 + 255) / 256;

    // layer 1: feat -> hcur
    gcn_gemm_scaled<<<gemm_blocks, 256, 0, stream>>>(feat, ns, W0, yproj, NNODES);
    gcn_zero_f32<<<zero_blocks, 256, 0, stream>>>(agg, NC / 4);
    gcn_spmm_scatter<<<spmm_blocks, 256, 0, stream>>>(yproj, src, dst, agg, NEDGES);
    gcn_finalize<<<fin_blocks, 256, 0, stream>>>(agg, nd, a0, hcur, hg + 0, NNODES);

    // layer 2: hcur -> hcur
    gcn_gemm_scaled<<<gemm_blocks, 256, 0, stream>>>(hcur, ns, W1, yproj, NNODES);
    gcn_zero_f32<<<zero_blocks, 256, 0, stream>>>(agg, NC / 4);
    gcn_spmm_scatter<<<spmm_blocks, 256, 0, stream>>>(yproj, src, dst, agg, NEDGES);
    gcn_finalize<<<fin_blocks, 256, 0, stream>>>(agg, nd, a1, hcur, hg + 128, NNODES);

    // layer 3: hcur -> out_h
    gcn_gemm_scaled<<<gemm_blocks, 256, 0, stream>>>(hcur, ns, W2, yproj, NNODES);
    gcn_zero_f32<<<zero_blocks, 256, 0, stream>>>(agg, NC / 4);
    gcn_spmm_scatter<<<spmm_blocks, 256, 0, stream>>>(yproj, src, dst, agg, NEDGES);
    gcn_finalize<<<fin_blocks, 256, 0, stream>>>(agg, nd, a2, out_h, hg + 256, NNODES);
}